// HighPass_18545668784580
// MI455X (gfx1250) — compile-verified
//
#include <hip/hip_runtime.h>
#include <stdint.h>

#define TILE_W 64
#define TILE_H 32
#define SWP 72            // padded LDS row stride (floats); interior at col 4 (16B aligned)
#define SH (TILE_H + 4)   // 36 rows (2-halo each side)
#define IMG 512
#define IMG2 (IMG * IMG)

// Output [32,1,512,512] fp32. Grid: (512/64, 512/32, 32), block 256 (= 8 wave32).
__global__ __launch_bounds__(256)
void highpass_tile_kernel(const float* __restrict__ src, float* __restrict__ out) {
    __shared__ float sS[SH * SWP];         // channel-summed tile + halo (10.1 KB)
    __shared__ float sO[TILE_H * TILE_W];  // output staging for async b128 stores (8 KB)

    const int tid = threadIdx.x;
    const int x0 = blockIdx.x * TILE_W;
    const int y0 = blockIdx.y * TILE_H;
    const int b  = blockIdx.z;
    const float* __restrict__ p0 = src + (size_t)b * 3u * IMG2;

    // ---- Stage 1a: interior columns via aligned float4 loads (16 chunks x 36 rows) ----
    // chunk e: row = e>>4 (0..35), col4 = (e&15)*4 within the 64-wide interior.
    for (int e = tid; e < 16 * SH; e += 256) {
        const int row  = e >> 4;
        const int col4 = (e & 15) << 2;
        int gy = y0 + row - 2;
        gy = (gy < 0) ? -gy : gy;
        gy = (gy > IMG - 1) ? (2 * (IMG - 1) - gy) : gy;
        const int gi = gy * IMG + x0 + col4;             // 16B aligned (x0 % 64 == 0)
        const float4 c0 = *(const float4*)(p0 + gi);
        const float4 c1 = *(const float4*)(p0 + gi + IMG2);
        const float4 c2 = *(const float4*)(p0 + gi + 2 * IMG2);
        float4 s;
        s.x = c0.x + c1.x + c2.x;
        s.y = c0.y + c1.y + c2.y;
        s.z = c0.z + c1.z + c2.z;
        s.w = c0.w + c1.w + c2.w;
        *(float4*)(&sS[row * SWP + 4 + col4]) = s;       // 16B aligned LDS store
    }
    // ---- Stage 1b: 4 halo columns (x-reflect) x 36 rows, scalar ----
    if (tid < 4 * SH) {
        const int row = tid >> 2;
        const int h   = tid & 3;                          // 0,1 -> left; 2,3 -> right
        int gx = x0 + ((h < 2) ? (h - 2) : (62 + h));     // x0-2, x0-1, x0+64, x0+65
        gx = (gx < 0) ? -gx : gx;
        gx = (gx > IMG - 1) ? (2 * (IMG - 1) - gx) : gx;
        int gy = y0 + row - 2;
        gy = (gy < 0) ? -gy : gy;
        gy = (gy > IMG - 1) ? (2 * (IMG - 1) - gy) : gy;
        const int gi  = gy * IMG + gx;
        const int scol = (h < 2) ? (2 + h) : (66 + h);    // LDS cols 2,3,68,69
        sS[row * SWP + scol] = p0[gi] + p0[gi + IMG2] + p0[gi + 2 * IMG2];
    }
    __syncthreads();

    // ---- Stage 2: each thread = one column x 8 rows, 5-row rolling window ----
    const int tx    = tid & (TILE_W - 1);   // 0..63
    const int ty    = tid >> 6;             // 0..3
    const int c     = tx + 4;               // column in sS coords (interior starts at 4)
    const int rbase = ty * (TILE_H / 4);    // first s-row of this thread's window

    const float SCALE = (float)(0.2989 + 0.587 + 0.114) / 7.0f;

    // a0 = h0 (center), a1 = h1 (x+-1 sum), a2 = h2 (x+-2 sum), for 5 rolling rows
    float a0[5], a1[5], a2[5];
#pragma unroll
    for (int i = 0; i < 4; ++i) {
        const float* rp = &sS[(rbase + i) * SWP + c];
        a0[i] = rp[0];
        a1[i] = rp[-1] + rp[1];
        a2[i] = rp[-2] + rp[2];
    }
#pragma unroll
    for (int r = 0; r < TILE_H / 4; ++r) {
        const float* rp = &sS[(rbase + r + 4) * SWP + c];
        a0[4] = rp[0];
        a1[4] = rp[-1] + rp[1];
        a2[4] = rp[-2] + rp[2];

        // vertical class sums (all 5 summed kernels are symmetric in both axes)
        const float u00 = a0[2],          u01 = a1[2],          u02 = a2[2];
        const float u10 = a0[1] + a0[3],  u11 = a1[1] + a1[3],  u12 = a2[1] + a2[3];
        const float u20 = a0[0] + a0[4],  u21 = a1[0] + a1[4],  u22 = a2[0] + a2[4];

        // F1sum == F3sum == 3x3 [[1,1,1],[1,-8,1],[1,1,1]]
        const float y1 = -8.f  * u00 +        u01 +        u10 +        u11;
        const float y2 = -24.f * u00 + 6.f  * u01 -        u02
                       + 6.f   * u10 + 3.f  * u11 -        u20;
        const float y4 = -16.f * u00 + 6.f  * u01 + 6.f  * u10 - 2.f * u11;
        const float y6 = -48.f * u00 + 24.f * u01 - 6.f  * u02
                       + 24.f  * u10 - 12.f * u11 + 4.f  * u12
                       - 6.f   * u20 + 4.f  * u21 + 2.f  * u22;
        const float y8 = -12.f * u00 + 8.f  * u01 - 2.f  * u02
                       + 8.f   * u10 - 6.f  * u11 + 2.f  * u12
                       - 2.f   * u20 + 2.f  * u21 -        u22;

        // groups: [F1,F2]/8, [F3(=F1),F4,F6]/4 with F6 weighted 2x; F8 raw
        const float v = floorf(y1 * 0.125f) + floorf(y2 * 0.125f)
                      + floorf(y1 * 0.25f)  + floorf(y4 * 0.25f)
                      + 2.0f * floorf(y6 * 0.25f) + y8;
        sO[(rbase + r) * TILE_W + tx] = v * SCALE;

        a0[0]=a0[1]; a0[1]=a0[2]; a0[2]=a0[3]; a0[3]=a0[4];
        a1[0]=a1[1]; a1[1]=a1[2]; a1[2]=a1[3]; a1[3]=a1[4];
        a2[0]=a2[1]; a2[1]=a2[2]; a2[2]=a2[3]; a2[3]=a2[4];
    }
    __syncthreads();

    // ---- Stage 3: async DMA the tile out of LDS, 16B per lane (CDNA5 async path) ----
    float* const ob = out + (size_t)b * IMG2 + (size_t)y0 * IMG + x0;
#pragma unroll
    for (int k = 0; k < 2; ++k) {
        const int q   = tid + k * 256;      // 0..511 chunks of 4 floats
        const int row = q >> 4;             // 16 chunks per 64-wide row
        const int c4  = (q & 15) << 2;
        const unsigned lofs  = (unsigned)(uint64_t)(uintptr_t)&sO[row * TILE_W + c4];
        const uint64_t gaddr = (uint64_t)(uintptr_t)(ob + row * IMG + c4);
        asm volatile("global_store_async_from_lds_b128 %0, %1, off"
                     :: "v"(gaddr), "v"(lofs) : "memory");
    }
#if __has_builtin(__builtin_amdgcn_s_wait_asynccnt)
    __builtin_amdgcn_s_wait_asynccnt(0);
#else
    asm volatile("s_wait_asynccnt 0" ::: "memory");
#endif
}

extern "C" void kernel_launch(void* const* d_in, const int* in_sizes, int n_in,
                              void* d_out, int out_size, void* d_ws, size_t ws_size,
                              hipStream_t stream) {
    (void)in_sizes; (void)n_in; (void)d_ws; (void)ws_size; (void)out_size;
    const float* src = (const float*)d_in[0];
    float* out = (float*)d_out;
    dim3 grid(IMG / TILE_W, IMG / TILE_H, 32);
    dim3 block(256, 1, 1);
    hipLaunchKernelGGL(highpass_tile_kernel, grid, block, 0, stream, src, out);
}